// loss_29291676959154
// MI455X (gfx1250) — compile-verified
//
#include <hip/hip_runtime.h>
#include <stdint.h>

// YOLO-v1 loss on gfx1250 (MI455X).
// Memory-bound streaming reduction: 176.6 MB in, 4 B out -> ~7.6 us HBM floor.
// Strategy: per block, async-copy 256 cells of pred (30 f32) + target (25 f32)
// into LDS with coalesced b128 async-to-LDS transfers, compute per-cell loss
// from LDS (bank-conflict-free AoS reads), wave32 shuffle reduce, one f64
// global atomic per block, tiny finalize kernel casts to f32.

#define SS 7
#define PRED_C 30   // 5*B + C
#define TGT_C 25    // 5 + C
#define BLOCK 256

__global__ void __launch_bounds__(1) zero_acc_kernel(double* acc) {
    acc[0] = 0.0;
}

__global__ void __launch_bounds__(1) finalize_kernel(const double* acc, float* out) {
    out[0] = (float)acc[0];
}

__global__ void __launch_bounds__(BLOCK)
yolo_loss_kernel(const float* __restrict__ pred,
                 const float* __restrict__ tgt,
                 long long cells,
                 double* __restrict__ acc)
{
    __shared__ float sp[BLOCK * PRED_C];   // 30720 B
    __shared__ float st[BLOCK * TGT_C];    // 25600 B
    __shared__ float swsum[BLOCK / 32];

    const int tid = threadIdx.x;
    const long long blockStart = (long long)blockIdx.x * BLOCK;
    long long rem = cells - blockStart;
    const int nCells = (rem < BLOCK) ? (int)rem : BLOCK;

    const float* gp = pred + blockStart * PRED_C;
    const float* gt = tgt  + blockStart * TGT_C;

    // ---- async global -> LDS staging (16B per lane per op, ASYNCcnt) ----
    const int nPred4 = (nCells * PRED_C + 3) >> 2;
    for (int k = tid; k < nPred4; k += BLOCK) {
        uint32_t laddr = (uint32_t)(uintptr_t)(&sp[k * 4]);
        uint64_t gaddr = (uint64_t)(uintptr_t)(gp + k * 4);
        asm volatile("global_load_async_to_lds_b128 %0, %1, off"
                     :: "v"(laddr), "v"(gaddr) : "memory");
    }
    const int nTgt4 = (nCells * TGT_C + 3) >> 2;
    for (int k = tid; k < nTgt4; k += BLOCK) {
        uint32_t laddr = (uint32_t)(uintptr_t)(&st[k * 4]);
        uint64_t gaddr = (uint64_t)(uintptr_t)(gt + k * 4);
        asm volatile("global_load_async_to_lds_b128 %0, %1, off"
                     :: "v"(laddr), "v"(gaddr) : "memory");
    }
    asm volatile("s_wait_asynccnt 0" ::: "memory");
    __syncthreads();

    // ---- per-cell loss ----
    float loss = 0.0f;
    if (tid < nCells) {
        const float* p = &sp[tid * PRED_C];
        const float* t = &st[tid * TGT_C];

        long long cell = blockStart + tid;
        const int j = (int)(cell % SS);          // col (axis 2)
        const int i = (int)((cell / SS) % SS);   // row (axis 1)
        const float colf = (float)j;
        const float rowf = (float)i;

        const float gx = t[0], gy = t[1], gw = t[2], gh = t[3], t4 = t[4];

        // ground-truth corner box (nb = 1)
        const float gcx = (gx + colf) / 7.0f;
        const float gcy = (gy + rowf) / 7.0f;
        const float gx1 = gcx - gw * 0.5f, gy1 = gcy - gh * 0.5f;
        const float gx2 = gcx + gw * 0.5f, gy2 = gcy + gh * 0.5f;
        const float ga  = fabsf((gx2 - gx1) * (gy2 - gy1));

        float iou[2];
        #pragma unroll
        for (int b = 0; b < 2; ++b) {
            const float bx = p[5 * b + 0], by = p[5 * b + 1];
            const float bw = p[5 * b + 2], bh = p[5 * b + 3];
            const float cx = (bx + colf) / 7.0f;
            const float cy = (by + rowf) / 7.0f;
            const float x1 = cx - bw * 0.5f, y1 = cy - bh * 0.5f;
            const float x2 = cx + bw * 0.5f, y2 = cy + bh * 0.5f;
            const float xx1 = fmaxf(x1, gx1), yy1 = fmaxf(y1, gy1);
            const float xx2 = fminf(x2, gx2), yy2 = fminf(y2, gy2);
            const float inter = fmaxf(xx2 - xx1, 0.0f) * fmaxf(yy2 - yy1, 0.0f);
            const float pa = fabsf((x2 - x1) * (y2 - y1));
            iou[b] = inter / (pa + ga - inter + 1e-8f);
        }
        // jnp.argmax: first max wins on ties
        const int bi = (iou[1] > iou[0]) ? 1 : 0;
        const float* sb = p + 5 * bi;
        const float px = sb[0], py = sb[1], pw = sb[2], ph = sb[3], pc = sb[4];

        // class loss: t4 * sum((pred_cls - tgt_cls)^2), summed at every cell
        float cls = 0.0f;
        #pragma unroll
        for (int c = 0; c < 20; ++c) {
            const float d = p[10 + c] - t[5 + c];
            cls += d * d;
        }
        loss = t4 * cls;

        if (t4 != 0.0f) {
            const float dx = px - gx, dy = py - gy;
            const float center = dx * dx + dy * dy;
            const float spw = (pw > 0.0f) ? 1.0f : ((pw < 0.0f) ? -1.0f : 0.0f);
            const float sph = (ph > 0.0f) ? 1.0f : ((ph < 0.0f) ? -1.0f : 0.0f);
            const float sw = spw * sqrtf(fabsf(pw) + 1e-8f) - sqrtf(gw);
            const float sh = sph * sqrtf(fabsf(ph) + 1e-8f) - sqrtf(gh);
            const float dim = sw * sw + sh * sh;
            const float dc = pc - t4;   // gc == t4 here
            loss += 5.0f * (center + dim) + dc * dc;
        } else {
            const float d0 = p[4] - t4;  // t4 == 0 here
            const float d1 = p[9] - t4;
            loss += 0.5f * (d0 * d0 + d1 * d1);
        }
    }

    // ---- wave32 reduce, then block reduce, then one f64 atomic ----
    #pragma unroll
    for (int off = 16; off > 0; off >>= 1)
        loss += __shfl_xor(loss, off, 32);
    if ((tid & 31) == 0) swsum[tid >> 5] = loss;
    __syncthreads();
    if (tid == 0) {
        float s = 0.0f;
        #pragma unroll
        for (int w = 0; w < BLOCK / 32; ++w) s += swsum[w];
        atomicAdd(acc, (double)s);   // global_atomic_add_f64
    }
}

extern "C" void kernel_launch(void* const* d_in, const int* in_sizes, int n_in,
                              void* d_out, int out_size, void* d_ws, size_t ws_size,
                              hipStream_t stream) {
    (void)n_in; (void)out_size; (void)ws_size;

    const float* pred = (const float*)d_in[0];   // [N,7,7,30] f32
    const float* tgt  = (const float*)d_in[1];   // [N,7,7,25] f32
    float* out = (float*)d_out;
    double* acc = (double*)d_ws;

    const long long cells = (long long)in_sizes[0] / PRED_C;  // N*49
    const int grid = (int)((cells + BLOCK - 1) / BLOCK);

    zero_acc_kernel<<<1, 1, 0, stream>>>(acc);
    yolo_loss_kernel<<<grid, BLOCK, 0, stream>>>(pred, tgt, cells, acc);
    finalize_kernel<<<1, 1, 0, stream>>>(acc, out);
}